// Con2D_self_30348238913579
// MI455X (gfx1250) — compile-verified
//
#include <hip/hip_runtime.h>

typedef __attribute__((ext_vector_type(2))) float v2f;
typedef __attribute__((ext_vector_type(8))) float v8f;

#define XW 4096           // input width/height
#define OW 4082           // output width/height (4096 - 15 + 1)
#define SX_ROWS 46        // 32 output rows + 14 halo
#define SX_COLS 80        // 64 output cols + 14 halo + 2 pad (A reads cols up to 79)
#define SX_STRIDE 82      // odd-ish stride (mod 64 banks = 18) -> conflict-free A reads

// Conv 15x15 as sum of Toeplitz GEMMs on f32 WMMA:
//   y[m,n] = sum_p sum_k A_p[m,k] * B_p[k,n]
//   A_p[m,k] = x[r0+m+p, c0+k]            (16 x 32, K padded 30->32)
//   B_p[k,n] = w[p, k-n] if 0<=k-n<15 else 0
__global__ __launch_bounds__(256) void conv15_wmma_f32(
    const float* __restrict__ x, const float* __restrict__ w,
    const float* __restrict__ b, float* __restrict__ out) {
  __shared__ float sx[SX_ROWS * SX_STRIDE];     // ~15 KB staged x tile (+halo)
  __shared__ float sB[15 * 8 * 32 * 2];         // 30 KB Toeplitz B fragments

  const int tid  = threadIdx.x;
  const int lane = tid & 31;                    // wave32
  const int wv   = tid >> 5;                    // 8 waves per block
  const int wr   = wv >> 2;                     // wave tile row (0..1)
  const int wc   = wv & 3;                      // wave tile col (0..3)

  const int blockR = blockIdx.y * 32;
  const int blockC = blockIdx.x * 64;

  // ---- stage x tile + halo into LDS (clamped at the image edge) ----
  for (int i = tid; i < SX_ROWS * SX_COLS; i += 256) {
    int lr = i / SX_COLS;
    int lc = i - lr * SX_COLS;
    int gr = blockR + lr; if (gr > XW - 1) gr = XW - 1;
    int gc = blockC + lc; if (gc > XW - 1) gc = XW - 1;
    sx[lr * SX_STRIDE + lc] = x[gr * XW + gc];
  }

  // ---- build Toeplitz B fragments in WMMA B-operand layout ----
  // sB[((p*8 + kq)*32 + lane)*2 + j] = B_p[4*kq + 2*(lane>>4) + j, lane&15]
  for (int i = tid; i < 15 * 8 * 32; i += 256) {
    int l  = i & 31;
    int kq = (i >> 5) & 7;
    int p  = i >> 8;
    int nn = l & 15;
    int kh = l >> 4;
#pragma unroll
    for (int j = 0; j < 2; ++j) {
      int k = 4 * kq + 2 * kh + j;
      int q = k - nn;
      sB[i * 2 + j] = (q >= 0 && q < 15) ? w[p * 15 + q] : 0.0f;
    }
  }
  __syncthreads();

  // ---- WMMA main loop: 15 p-slices x 8 K-steps of v_wmma_f32_16x16x4_f32 ----
  const int n  = lane & 15;                     // output column within tile / A row m
  const int kh = lane >> 4;                     // K-half select
  const int arow0 = wr * 16 + n;                // A row m = lane&15
  const int acol0 = wc * 16 + 2 * kh;           // A col base for this K-half

  v8f acc = {};
  for (int p = 0; p < 15; ++p) {
    const float* arow = &sx[(arow0 + p) * SX_STRIDE + acol0];
    const float* brow = &sB[p * 512 + lane * 2];
#pragma unroll
    for (int kq = 0; kq < 8; ++kq) {
      v2f a  = *(const v2f*)(arow + 4 * kq);    // A frag: K = {4kq+2kh, 4kq+2kh+1}
      v2f bb = *(const v2f*)(brow + kq * 64);   // B frag (precomputed layout)
      acc = __builtin_amdgcn_wmma_f32_16x16x4_f32(
          /*neg_a=*/false, a, /*neg_b=*/false, bb,
          /*c_mod=*/(short)0, acc, /*reuse_a=*/false, /*reuse_b=*/false);
    }
  }

  // ---- store C/D: VGPR i -> row i + 8*kh, col n ----
  const float bias = b[0];
  const int r0 = blockR + wr * 16 + 8 * kh;
  const int c0 = blockC + wc * 16 + n;
  if (c0 < OW) {
#pragma unroll
    for (int i = 0; i < 8; ++i) {
      int r = r0 + i;
      if (r < OW) out[r * OW + c0] = acc[i] + bias;
    }
  }
}

extern "C" void kernel_launch(void* const* d_in, const int* in_sizes, int n_in,
                              void* d_out, int out_size, void* d_ws, size_t ws_size,
                              hipStream_t stream) {
  (void)in_sizes; (void)n_in; (void)out_size; (void)d_ws; (void)ws_size;
  const float* x = (const float*)d_in[0];   // 4096*4096
  const float* w = (const float*)d_in[1];   // 15*15
  const float* b = (const float*)d_in[2];   // 1
  float* out = (float*)d_out;               // 4082*4082

  dim3 grid(64, 128);                       // 64*64 cols, 128*32 rows cover 4082
  conv15_wmma_f32<<<grid, 256, 0, stream>>>(x, w, b, out);
}